// SinkhornOTLoss_8942121910865
// MI455X (gfx1250) — compile-verified
//
#include <hip/hip_runtime.h>

// Sinkhorn OT loss on MI455X (gfx1250).
//
// Design (see analysis):
//  - K = exp(-20*M) materialized once in workspace (128 MB, fits in 192 MB L2).
//  - Up to 100 iterations, each = column pass (v = b/(K^T u + eps)) + row pass
//    (u = a/(K v + eps)); error check passes after iters 1 and 51 set a device
//    `stop` flag; all later kernels no-op on it (graph-capture safe early exit).
//  - Column/error passes use V_WMMA_F32_16X16X4_F32 with a replicated-u A
//    operand: D[i,j] = sum_k u[k] * K[k, m0+j]  (all rows of D identical).
//  - Row pass + final loss pass are coalesced float4 streaming reductions.

typedef __attribute__((ext_vector_type(2))) float v2f;
typedef __attribute__((ext_vector_type(8))) float v8f;

#define N_TOP   512
#define M_VOC   65536
#define ALPHA   20.0f
#define A_MARG  (1.0f/512.0f)
#define B_MARG  (1.0f/65536.0f)
#define EPSV    1e-16f
#define STOPTHR 0.005f

// ---------------------------------------------------------------- init ------
// K = exp(-alpha*M), u = 1/n, stop = 0
__global__ void __launch_bounds__(256) sink_init(const float* __restrict__ Mm,
                                                 float* __restrict__ K,
                                                 float* __restrict__ u,
                                                 int* __restrict__ stop) {
    size_t gid    = (size_t)blockIdx.x * blockDim.x + threadIdx.x;
    size_t stride = (size_t)gridDim.x * blockDim.x;
    size_t nvec   = (size_t)N_TOP * M_VOC / 4;
    const float4* M4 = (const float4*)Mm;
    float4*       K4 = (float4*)K;
    for (size_t i = gid; i < nvec; i += stride) {
        float4 m = M4[i];
        float4 k;
        k.x = __expf(-ALPHA * m.x);
        k.y = __expf(-ALPHA * m.y);
        k.z = __expf(-ALPHA * m.z);
        k.w = __expf(-ALPHA * m.w);
        K4[i] = k;
    }
    if (gid < N_TOP) u[gid] = A_MARG;
    if (gid == 0)    *stop  = 0;
}

// --------------------------------------------------- column pass (WMMA) -----
// Computes per-wave, for 16 columns m0..m0+15:  ktu[j] = sum_n K[n, m0+j]*u[n]
// A (16x4 f32): A[i,k] = u[k0+k] replicated over rows i.
//   Layout: V0 = K=0 (lanes 0-15) / K=2 (lanes 16-31); V1 = K=1 / K=3.
// B (4x16 f32): rows striped across lanes: lanes 0-15 -> K=0 (V0), K=1 (V1);
//   lanes 16-31 -> K=2 (V0), K=3 (V1); column N = lane & 15.
// C/D: V0 lanes 0-15 = row M=0, cols N=0..15 (all rows identical here).
__device__ __forceinline__ v8f ktu_wave(const float* __restrict__ K,
                                        const float* __restrict__ su,
                                        int m0, int half, int l) {
    v8f c = {};
    const float* kcol = K + (size_t)(2 * half) * M_VOC + (m0 + l);
#pragma unroll 4
    for (int k0 = 0; k0 < N_TOP; k0 += 4) {
        v2f A, B;
        A.x = su[k0 + 2 * half];
        A.y = su[k0 + 2 * half + 1];
        const float* kp = kcol + (size_t)k0 * M_VOC;
        B.x = kp[0];
        B.y = kp[M_VOC];
        c = __builtin_amdgcn_wmma_f32_16x16x4_f32(false, A, false, B,
                                                  (short)0, c, false, false);
    }
    return c;
}

// v[m] = b / (K^T u + eps).  8 waves/block -> 128 columns/block, 512 blocks.
__global__ void __launch_bounds__(256) sink_colpass(const float* __restrict__ K,
                                                    const float* __restrict__ u,
                                                    float* __restrict__ v,
                                                    const int* __restrict__ stop) {
    if (*stop) return;
    __shared__ float su[N_TOP];
    for (int i = threadIdx.x; i < N_TOP; i += 256) su[i] = u[i];
    __syncthreads();
    const int lane = threadIdx.x & 31;
    const int wave = threadIdx.x >> 5;
    const int half = lane >> 4;
    const int l    = lane & 15;
    const int m0   = (blockIdx.x * 8 + wave) * 16;
    v8f c = ktu_wave(K, su, m0, half, l);
    if (half == 0) v[m0 + l] = B_MARG / (c[0] + EPSV);
}

// Error pass: partial[block] = sum_m |v[m]*(K^T u)[m] - b| over its columns.
__global__ void __launch_bounds__(256) sink_errpass(const float* __restrict__ K,
                                                    const float* __restrict__ u,
                                                    const float* __restrict__ v,
                                                    float* __restrict__ partials,
                                                    const int* __restrict__ stop) {
    if (*stop) return;
    __shared__ float su[N_TOP];
    __shared__ float sred[256];
    for (int i = threadIdx.x; i < N_TOP; i += 256) su[i] = u[i];
    __syncthreads();
    const int lane = threadIdx.x & 31;
    const int wave = threadIdx.x >> 5;
    const int half = lane >> 4;
    const int l    = lane & 15;
    const int m0   = (blockIdx.x * 8 + wave) * 16;
    v8f c = ktu_wave(K, su, m0, half, l);
    float val = 0.0f;
    if (half == 0) val = fabsf(v[m0 + l] * c[0] - B_MARG);
    sred[threadIdx.x] = val;
    __syncthreads();
    for (int s = 128; s > 0; s >>= 1) {
        if (threadIdx.x < s) sred[threadIdx.x] += sred[threadIdx.x + s];
        __syncthreads();
    }
    if (threadIdx.x == 0) partials[blockIdx.x] = sred[0];
}

// Deterministic final error reduce; sets stop flag if converged.
__global__ void __launch_bounds__(512) sink_errreduce(const float* __restrict__ partials,
                                                      int* __restrict__ stop) {
    if (*stop) return;
    __shared__ float s[512];
    const int t = threadIdx.x;
    s[t] = partials[t];
    __syncthreads();
    for (int k = 256; k > 0; k >>= 1) {
        if (t < k) s[t] += s[t + k];
        __syncthreads();
    }
    if (t == 0 && s[0] <= STOPTHR) *stop = 1;
}

// ------------------------------------------------------------- row pass -----
// u[n] = a / (sum_m K[n,m]*v[m] + eps).  One row per block, coalesced float4.
__global__ void __launch_bounds__(256) sink_rowpass(const float* __restrict__ K,
                                                    const float* __restrict__ v,
                                                    float* __restrict__ u,
                                                    const int* __restrict__ stop) {
    if (*stop) return;
    const int row = blockIdx.x;
    const float4* Kr = (const float4*)(K + (size_t)row * M_VOC);
    const float4* V4 = (const float4*)v;
    float acc = 0.0f;
    for (int i = threadIdx.x; i < M_VOC / 4; i += 256) {
        float4 k4 = Kr[i];
        float4 v4 = V4[i];
        acc = fmaf(k4.x, v4.x, acc);
        acc = fmaf(k4.y, v4.y, acc);
        acc = fmaf(k4.z, v4.z, acc);
        acc = fmaf(k4.w, v4.w, acc);
    }
    __shared__ float sred[256];
    sred[threadIdx.x] = acc;
    __syncthreads();
    for (int s = 128; s > 0; s >>= 1) {
        if (threadIdx.x < s) sred[threadIdx.x] += sred[threadIdx.x + s];
        __syncthreads();
    }
    if (threadIdx.x == 0) u[row] = A_MARG / (sred[0] + EPSV);
}

// ------------------------------------------------------------ loss pass -----
// partial[row] = u[row] * sum_m K[row,m]*v[m]*M[row,m]
__global__ void __launch_bounds__(256) sink_losspass(const float* __restrict__ Mm,
                                                     const float* __restrict__ K,
                                                     const float* __restrict__ u,
                                                     const float* __restrict__ v,
                                                     float* __restrict__ partials) {
    const int row = blockIdx.x;
    const float4* Kr = (const float4*)(K  + (size_t)row * M_VOC);
    const float4* Mr = (const float4*)(Mm + (size_t)row * M_VOC);
    const float4* V4 = (const float4*)v;
    float acc = 0.0f;
    for (int i = threadIdx.x; i < M_VOC / 4; i += 256) {
        float4 k4 = Kr[i];
        float4 m4 = Mr[i];
        float4 v4 = V4[i];
        acc = fmaf(k4.x * v4.x, m4.x, acc);
        acc = fmaf(k4.y * v4.y, m4.y, acc);
        acc = fmaf(k4.z * v4.z, m4.z, acc);
        acc = fmaf(k4.w * v4.w, m4.w, acc);
    }
    __shared__ float sred[256];
    sred[threadIdx.x] = acc;
    __syncthreads();
    for (int s = 128; s > 0; s >>= 1) {
        if (threadIdx.x < s) sred[threadIdx.x] += sred[threadIdx.x + s];
        __syncthreads();
    }
    if (threadIdx.x == 0) partials[row] = u[row] * sred[0];
}

__global__ void __launch_bounds__(512) sink_finalize(const float* __restrict__ partials,
                                                     float* __restrict__ out) {
    __shared__ float s[512];
    const int t = threadIdx.x;
    s[t] = partials[t];
    __syncthreads();
    for (int k = 256; k > 0; k >>= 1) {
        if (t < k) s[t] += s[t + k];
        __syncthreads();
    }
    if (t == 0) out[0] = s[0] * 100.0f;  // WEIGHT_LOSS_ECR
}

// ---------------------------------------------------------------- launch ----
extern "C" void kernel_launch(void* const* d_in, const int* in_sizes, int n_in,
                              void* d_out, int out_size, void* d_ws, size_t ws_size,
                              hipStream_t stream) {
    const float* Mm = (const float*)d_in[0];

    // Workspace layout (~128.3 MB):
    float* K        = (float*)d_ws;
    float* u        = K + (size_t)N_TOP * M_VOC;
    float* v        = u + N_TOP;
    float* partials = v + M_VOC;
    int*   stop     = (int*)(partials + 512);

    float* out = (float*)d_out;

    sink_init<<<4096, 256, 0, stream>>>(Mm, K, u, stop);

    for (int cpt = 1; cpt <= 100; ++cpt) {
        sink_colpass<<<512, 256, 0, stream>>>(K, u, v, stop);
        sink_rowpass<<<512, 256, 0, stream>>>(K, v, u, stop);
        if (cpt == 1 || cpt == 51) {
            // err = sum |v * (K^T u_new) - b|, matching the reference's
            // check at cpt % 50 == 1.
            sink_errpass<<<512, 256, 0, stream>>>(K, u, v, partials, stop);
            sink_errreduce<<<1, 512, 0, stream>>>(partials, stop);
        }
    }

    sink_losspass<<<512, 256, 0, stream>>>(Mm, K, u, v, partials);
    sink_finalize<<<1, 512, 0, stream>>>(partials, out);
}